// ConvertParamsLayer_76596446757534
// MI455X (gfx1250) — compile-verified
//
#include <hip/hip_runtime.h>

typedef __attribute__((ext_vector_type(2))) float v2f;
typedef __attribute__((ext_vector_type(8))) float v8f;

#define B_  64
#define NH  1024
#define NV  2048

// ---------------------------------------------------------------------------
// Prep: sc[b*NH+h] = { scale = sqrt(varh1/varh2), coeff = muh1 - scale*muh2 }
// ---------------------------------------------------------------------------
__global__ __launch_bounds__(256) void prep_sc_kernel(
    const float* __restrict__ muh1, const float* __restrict__ muh2,
    const float* __restrict__ varh1, const float* __restrict__ varh2,
    float2* __restrict__ sc) {
  int i = blockIdx.x * blockDim.x + threadIdx.x;
  if (i < B_ * NH) {
    float s = sqrtf(varh1[i] / varh2[i]);   // == sqrt(v1)/sqrt(v2), v>0
    float2 p;
    p.x = s;
    p.y = muh1[i] - s * muh2[i];
    sc[i] = p;
  }
}

// ---------------------------------------------------------------------------
// Main: one wave32 per (batch b, 16-wide v strip).  Streams wt1 once:
//   - NT loads of wt1 directly in WMMA f32 16x16x4 A-matrix layout
//     (lane = v0 + (lane&15); vgpr0 -> K = 2*(lane>>4), vgpr1 -> K+1)
//   - NT stores wt2 = scale * wt1 from the same registers
//   - B-matrix = coeff[h0+k] broadcast -> every D column = GEMV partials
// Accumulates D over all 1024 h (two interleaved accumulators for ILP),
// then extracts the 16 per-v partials and writes b2 = b1 + partial.
// ---------------------------------------------------------------------------
__global__ __launch_bounds__(256) void convert_params_stream_kernel(
    const float* __restrict__ wt1, const float* __restrict__ b1,
    const float2* __restrict__ sc,
    float* __restrict__ b2, float* __restrict__ wt2) {
  const int lane = threadIdx.x & 31;
  const int wave = threadIdx.x >> 5;
  const int wg   = blockIdx.x * 8 + wave;   // 0 .. 8191
  const int b    = wg >> 7;                 // / (NV/16)
  const int v0   = (wg & 127) << 4;         // v strip base
  const int half = lane >> 4;               // 0: lanes 0-15, 1: lanes 16-31
  const int m    = lane & 15;               // A-matrix row -> v0 + m
  const int k0   = half << 1;               // vgpr0 holds K = 0 / 2
  const int k1   = k0 + 1;                  // vgpr1 holds K = 1 / 3

  const size_t matBase = (size_t)b * (size_t)NH * (size_t)NV;
  const float* p0 = wt1 + matBase + (size_t)k0 * NV + (v0 + m);
  const float* p1 = wt1 + matBase + (size_t)k1 * NV + (v0 + m);
  float*       q0 = wt2 + matBase + (size_t)k0 * NV + (v0 + m);
  float*       q1 = wt2 + matBase + (size_t)k1 * NV + (v0 + m);
  const float2* s0p = sc + (size_t)b * NH + k0;
  const float2* s1p = sc + (size_t)b * NH + k1;

  v8f acc0 = {};
  v8f acc1 = {};

  const ptrdiff_t stepW = 4 * (ptrdiff_t)NV;  // advance one h-quad

#pragma unroll 4
  for (int it = 0; it < NH / 8; ++it) {
    // ---- h-quad 0 -> acc0 ----
    {
      float x0 = __builtin_nontemporal_load(p0);
      float x1 = __builtin_nontemporal_load(p1);
      float2 s0 = *s0p;
      float2 s1 = *s1p;
      __builtin_nontemporal_store(x0 * s0.x, q0);
      __builtin_nontemporal_store(x1 * s1.x, q1);
      v2f a  = {x0, x1};
      v2f bb = {s0.y, s1.y};
      acc0 = __builtin_amdgcn_wmma_f32_16x16x4_f32(
          false, a, false, bb, (short)0, acc0, false, false);
      p0 += stepW; p1 += stepW; q0 += stepW; q1 += stepW;
      s0p += 4; s1p += 4;
    }
    // ---- h-quad 1 -> acc1 ----
    {
      float x0 = __builtin_nontemporal_load(p0);
      float x1 = __builtin_nontemporal_load(p1);
      float2 s0 = *s0p;
      float2 s1 = *s1p;
      __builtin_nontemporal_store(x0 * s0.x, q0);
      __builtin_nontemporal_store(x1 * s1.x, q1);
      v2f a  = {x0, x1};
      v2f bb = {s0.y, s1.y};
      acc1 = __builtin_amdgcn_wmma_f32_16x16x4_f32(
          false, a, false, bb, (short)0, acc1, false, false);
      p0 += stepW; p1 += stepW; q0 += stepW; q1 += stepW;
      s0p += 4; s1p += 4;
    }
  }

  v8f acc = acc0 + acc1;

  // D layout: vgpr r -> M=r (lanes 0-15) / M=8+r (lanes 16-31); all N equal.
  // lanes 0-7 write v0+0..7, lanes 16-23 write v0+8..15.
  if (m < 8) {
    float val = 0.0f;
#pragma unroll
    for (int j = 0; j < 8; ++j) val = (j == m) ? acc[j] : val;
    size_t idx = (size_t)b * NV + (v0 + half * 8 + m);
    b2[idx] = b1[idx] + val;
  }
}

// ---------------------------------------------------------------------------
extern "C" void kernel_launch(void* const* d_in, const int* in_sizes, int n_in,
                              void* d_out, int out_size, void* d_ws, size_t ws_size,
                              hipStream_t stream) {
  const float* b1    = (const float*)d_in[0];
  const float* wt1   = (const float*)d_in[1];
  const float* muh1  = (const float*)d_in[2];
  const float* muh2  = (const float*)d_in[3];
  const float* varh1 = (const float*)d_in[4];
  const float* varh2 = (const float*)d_in[5];

  float* b2  = (float*)d_out;                       // [B, NV]
  float* wt2 = (float*)d_out + (size_t)B_ * NV;     // [B, NH, NV]
  float2* sc = (float2*)d_ws;                       // 512 KB scratch

  prep_sc_kernel<<<(B_ * NH + 255) / 256, 256, 0, stream>>>(
      muh1, muh2, varh1, varh2, sc);

  // 8192 waves: 64 batches * 128 v-strips; 8 waves per 256-thread block.
  convert_params_stream_kernel<<<(B_ * (NV / 16)) / 8, 256, 0, stream>>>(
      wt1, b1, sc, b2, wt2);
}